// PatternMemory_45681272160870
// MI455X (gfx1250) — compile-verified
//
#include <hip/hip_runtime.h>
#include <math.h>

// ---------------------------------------------------------------------------
// PatternMemory for MI455X (gfx1250, wave32).
//  Kernel 1: fp32 WMMA (v_wmma_f32_16x16x4_f32) GEMM  proj = E @ W^T + b,
//            hash codes via __ballot; A-tile staged to LDS through the
//            gfx1250 async-to-LDS path (GLOBAL_LOAD_ASYNC_TO_LDS_B128).
//  Kernel 2: one wave32 per query hierarchical centroid walk (data-dependent
//            gather -> VALU FMAs + shfl_xor reductions, not a WMMA shape).
// ---------------------------------------------------------------------------

typedef __attribute__((ext_vector_type(2))) float v2f;
typedef __attribute__((ext_vector_type(4))) float v4f;
typedef __attribute__((ext_vector_type(8))) float v8f;
typedef int v4i __attribute__((vector_size(16)));   // builtin's pointee type

#define DIM        512
#define NTH        128          // T*H = 8*16 output columns
#define LDS_STRIDE 516          // 512 + 4 pad floats: 16-lane row stride hits
                                // distinct LDS banks for b64 reads

#if __has_builtin(__builtin_amdgcn_global_load_async_to_lds_b128) && \
    __has_builtin(__builtin_amdgcn_s_wait_asynccnt)
#define HAVE_ASYNC_LDS 1
#else
#define HAVE_ASYNC_LDS 0
#endif

// blockDim = (32, 8): 8 waves, each wave owns one 16x16 tile (one hash table).
__global__ __launch_bounds__(256)
void proj_hash_kernel(const float* __restrict__ E, const float* __restrict__ W,
                      const float* __restrict__ Hb,
                      float* __restrict__ out_codes, float* __restrict__ out_proj) {
    __shared__ float As[16 * LDS_STRIDE];

    const int lane  = threadIdx.x;          // 0..31
    const int wv    = threadIdx.y;          // 0..7 == table index t == n-tile
    const int tid   = wv * 32 + lane;       // 0..255
    const int qbase = blockIdx.x * 16;

    // Cooperatively stage the 16x512 fp32 A-tile (32 KB) into LDS once;
    // reused by all 8 waves (cuts A global traffic 8x).
#if HAVE_ASYNC_LDS
    #pragma unroll
    for (int i = 0; i < 8; ++i) {
        int idx = tid + i * 256;            // 2048 float4 slots
        int row = idx >> 7;                 // 0..15
        int c4  = idx & 127;                // 0..127
        const float* g = E + (size_t)(qbase + row) * DIM + c4 * 4;
        float*       l = &As[row * LDS_STRIDE + c4 * 4];
        __builtin_amdgcn_global_load_async_to_lds_b128(
            (v4i*)g, (v4i*)l, 0, 0);
    }
    __builtin_amdgcn_s_wait_asynccnt(0);
    __syncthreads();
#else
    #pragma unroll
    for (int i = 0; i < 8; ++i) {
        int idx = tid + i * 256;            // 2048 float4 slots
        int row = idx >> 7;                 // 0..15
        int c4  = idx & 127;                // 0..127
        v4f v = *(const v4f*)(E + (size_t)(qbase + row) * DIM + c4 * 4);
        *(v4f*)(&As[row * LDS_STRIDE + c4 * 4]) = v;
    }
    __syncthreads();
#endif

    const int n     = lane & 15;            // column within tile == hash bit h
    const int half  = lane >> 4;            // K sub-offset selector
    const int nbase = wv * 16;
    const float* __restrict__ wrow = W + (size_t)(nbase + n) * DIM;
    const float* __restrict__ arow = &As[n * LDS_STRIDE];   // m = lane&15

    v8f acc = {0.f, 0.f, 0.f, 0.f, 0.f, 0.f, 0.f, 0.f};

    // K loop: 128 x V_WMMA_F32_16X16X4_F32 (exact fp32 semantics).
    for (int k = 0; k < DIM; k += 4) {
        v2f a = *(const v2f*)(arow + k + 2 * half);   // ds_load_b64
        v2f b = *(const v2f*)(wrow + k + 2 * half);   // global_load_b64 (L2 hit)
        // locality 3 -> WGP-scope prefetch (pulls into all cache levels)
        __builtin_prefetch(wrow + (((k + 64) & (DIM - 1)) + 2 * half), 0, 3);
        // (neg_a, A, neg_b, B, c_mod, C, reuse_a, reuse_b)
        acc = __builtin_amdgcn_wmma_f32_16x16x4_f32(
            false, a, false, b, (short)0, acc, false, false);
    }

    const float bias = Hb[nbase + n];

    // D layout: acc[r] -> row (r + 8*half), col n. A row's 16 sign bits live
    // in one 16-lane half of the wave -> one __ballot gives the hash code.
    #pragma unroll
    for (int r = 0; r < 8; ++r) {
        float v = acc[r] + bias;
        int   m = r + 8 * half;
        out_proj[(size_t)(qbase + m) * NTH + nbase + n] = v;
        unsigned long long bal = __ballot(v > 0.0f);
        if (lane == 0)
            out_codes[(size_t)(qbase + r) * 8 + wv] =
                (float)(unsigned)(bal & 0xFFFFull);
        if (lane == 16)
            out_codes[(size_t)(qbase + r + 8) * 8 + wv] =
                (float)(unsigned)((bal >> 16) & 0xFFFFull);
    }
}

// One wave32 per query; blockDim 256 -> 8 queries/block.
__global__ __launch_bounds__(256)
void cluster_kernel(const float* __restrict__ E, const float* __restrict__ C0,
                    const float* __restrict__ C1, const float* __restrict__ C2,
                    float* __restrict__ out_leaf, float* __restrict__ out_path,
                    float* __restrict__ out_mind) {
    const int lane = threadIdx.x & 31;
    const int q    = blockIdx.x * 8 + (threadIdx.x >> 5);

    float e[16];
    #pragma unroll
    for (int j = 0; j < 16; ++j)
        e[j] = E[(size_t)q * DIM + j * 32 + lane];

    const float* levels[3] = {C0, C1, C2};
    int current = 0;

    #pragma unroll
    for (int l = 0; l < 3; ++l) {
        const int start = current * 10;                 // BRANCH = 10
        float best = 3.402823466e38f;
        int   bestc = 0;
        for (int c = 0; c < 10; ++c) {
            const float* __restrict__ cv = levels[l] + (size_t)(start + c) * DIM;
            float s = 0.f;
            #pragma unroll
            for (int j = 0; j < 16; ++j) {
                float d = cv[j * 32 + lane] - e[j];
                s = fmaf(d, d, s);
            }
            #pragma unroll
            for (int off = 16; off > 0; off >>= 1)
                s += __shfl_xor(s, off, 32);            // wave32 reduction
            if (s < best) { best = s; bestc = c; }      // strict < == first-min argmin
        }
        current = start + bestc;
        if (lane == 0) {
            out_path[q * 3 + l] = (float)current;
            out_mind[q * 3 + l] = sqrtf(best);          // sqrt after: monotone
        }
    }
    if (lane == 0) out_leaf[q] = (float)current;
}

extern "C" void kernel_launch(void* const* d_in, const int* in_sizes, int n_in,
                              void* d_out, int out_size, void* d_ws, size_t ws_size,
                              hipStream_t stream) {
    (void)n_in; (void)out_size; (void)d_ws; (void)ws_size;
    const float* E  = (const float*)d_in[0];   // [Q,512]
    const float* W  = (const float*)d_in[1];   // [8,16,512] -> [128,512]
    const float* Hb = (const float*)d_in[2];   // [128]
    const float* C0 = (const float*)d_in[3];   // [10,512]
    const float* C1 = (const float*)d_in[4];   // [100,512]
    const float* C2 = (const float*)d_in[5];   // [1000,512]

    const int Q = in_sizes[0] / DIM;           // 16384

    // Outputs concatenated flat in reference return order (all as float):
    float* out       = (float*)d_out;
    float* out_codes = out;                          // [Q,8]
    float* out_leaf  = out_codes + (size_t)Q * 8;    // [Q]
    float* out_path  = out_leaf + Q;                 // [Q,3]
    float* out_mind  = out_path + (size_t)Q * 3;     // [Q,3]
    float* out_proj  = out_mind + (size_t)Q * 3;     // [Q,8,16]

    dim3 b1(32, 8);
    proj_hash_kernel<<<Q / 16, b1, 0, stream>>>(E, W, Hb, out_codes, out_proj);
    cluster_kernel<<<Q / 8, 256, 0, stream>>>(E, C0, C1, C2,
                                              out_leaf, out_path, out_mind);
}